// GATNet_63608465654164
// MI455X (gfx1250) — compile-verified
//
#include <hip/hip_runtime.h>
#include <hip/hip_bf16.h>
#include <cstdint>
#include <cstddef>

typedef float v2f __attribute__((ext_vector_type(2)));
typedef float v8f __attribute__((ext_vector_type(8)));

#define NEG_SLOPE 0.2f
#define GAT_EPS   1e-16f

// ---------------------------------------------------------------------------
// float atomic max via orderable-int trick (lowers to global_atomic_max/min)
// ---------------------------------------------------------------------------
__device__ __forceinline__ void atomicMaxF32(float* addr, float val) {
    if (val >= 0.0f) {
        atomicMax((int*)addr, __float_as_int(val));
    } else {
        atomicMin((unsigned int*)addr, (unsigned int)__float_as_int(val));
    }
}

// ---------------------------------------------------------------------------
// fills / init
// ---------------------------------------------------------------------------
__global__ void fill_f32(float* __restrict__ p, float v, long long n) {
    long long i = (long long)blockIdx.x * blockDim.x + threadIdx.x;
    if (i < n) p[i] = v;
}

__global__ void init_out_bias(float* __restrict__ out, const float* __restrict__ b2, int N) {
    int i = blockIdx.x * blockDim.x + threadIdx.x;
    if (i < 2 * N) out[i] = b2[i & 1];
}

// ---------------------------------------------------------------------------
// Layer-1 projection: h1[N,128] = x[N,128] @ W1[128,128], fp32 WMMA 16x16x4.
// Block = 256 threads = 8 waves; block -> 16-row stripe, wave w -> cols [16w,16w+16)
// A frag (16x4, 32-bit): lane L: m=L&15, k = (L>>4)*2 + {0,1}   (2 contiguous f32)
// B frag (4x16, 32-bit): lane L: n=L&15, k = v + (L>>4)*2       (v = vgpr index)
// C/D (16x16, 32-bit):   lane L: n=L&15, m = v + (L>=16 ? 8:0)
// ---------------------------------------------------------------------------
__global__ void __launch_bounds__(256) gemm1_wmma(const float* __restrict__ x,
                                                  const float* __restrict__ W,
                                                  float* __restrict__ h1, int N) {
    const int lane = threadIdx.x & 31;
    const int wave = threadIdx.x >> 5;           // 0..7 column tile
    const int row0 = blockIdx.x * 16;
    if (row0 + 15 >= N && row0 >= N) return;     // N is a multiple of 16 here
    const int col0 = wave * 16;
    const int mA   = lane & 15;
    const int kHi  = (lane >> 4) * 2;            // 0 or 2

    v8f c = {};
    const float* arow = x + (size_t)(row0 + mA) * 128 + kHi;
    #pragma unroll 4
    for (int kk = 0; kk < 128; kk += 4) {
        v2f a, b;
        a.x = arow[kk + 0];
        a.y = arow[kk + 1];
        const float* bp = W + (size_t)(kk + kHi) * 128 + col0 + mA;
        b.x = bp[0];
        b.y = bp[128];
        c = __builtin_amdgcn_wmma_f32_16x16x4_f32(false, a, false, b,
                                                  (short)0, c, false, false);
    }
    const int mOut = (lane >> 4) * 8;
    const int nOut = lane & 15;
    float* op = h1 + (size_t)(row0 + mOut) * 128 + col0 + nOut;
    #pragma unroll
    for (int v = 0; v < 8; ++v) op[(size_t)v * 128] = c[v];
}

// ---------------------------------------------------------------------------
// Per-node attention logits, layer 1: a_src[n,h] = dot(h1[n,h,:], att_src[h,:])
// One wave per node; lanes 0..15 -> head 0, lanes 16..31 -> head 1.
// ---------------------------------------------------------------------------
__global__ void node_att1(const float* __restrict__ h1,
                          const float* __restrict__ att_src,
                          const float* __restrict__ att_dst,
                          float* __restrict__ a_src, float* __restrict__ a_dst, int N) {
    int node = blockIdx.x * (blockDim.x >> 5) + (threadIdx.x >> 5);
    if (node >= N) return;
    int lane = threadIdx.x & 31;
    float4 hv = ((const float4*)(h1 + (size_t)node * 128))[lane];
    float4 av = ((const float4*)att_src)[lane];
    float4 dv = ((const float4*)att_dst)[lane];
    float ps = hv.x * av.x + hv.y * av.y + hv.z * av.z + hv.w * av.w;
    float pd = hv.x * dv.x + hv.y * dv.y + hv.z * dv.z + hv.w * dv.w;
    #pragma unroll
    for (int m = 1; m <= 8; m <<= 1) {           // reduce within 16-lane head groups
        ps += __shfl_xor(ps, m, 32);
        pd += __shfl_xor(pd, m, 32);
    }
    if ((lane & 15) == 0) {
        int hd = lane >> 4;
        a_src[(size_t)node * 2 + hd] = ps;
        a_dst[(size_t)node * 2 + hd] = pd;
    }
}

__device__ __forceinline__ void edge_ids(const int* __restrict__ ei, long long e,
                                         int E, int* src, int* dst) {
    if (e < (long long)E) { *src = ei[e]; *dst = ei[(long long)E + e]; }
    else                  { *src = *dst = (int)(e - E); }
}

// ---------------------------------------------------------------------------
// Edge pass 1 (layer 1, H=2): e = leaky_relu(a_src[src]+a_dst[dst]); seg-max
// ---------------------------------------------------------------------------
__global__ void edge_max1(const float* __restrict__ a_src, const float* __restrict__ a_dst,
                          const int* __restrict__ ei, float* __restrict__ ebuf,
                          float* __restrict__ mbuf, int E, int N) {
    long long e = (long long)blockIdx.x * blockDim.x + threadIdx.x;
    if (e >= (long long)E + N) return;
    int src, dst; edge_ids(ei, e, E, &src, &dst);
    #pragma unroll
    for (int hh = 0; hh < 2; ++hh) {
        float v = a_src[(size_t)src * 2 + hh] + a_dst[(size_t)dst * 2 + hh];
        v = v > 0.0f ? v : NEG_SLOPE * v;
        ebuf[e * 2 + hh] = v;
        atomicMaxF32(&mbuf[(size_t)dst * 2 + hh], v);
    }
}

// Edge pass 2 (layer 1): e <- exp(e - m[dst]); seg-sum
__global__ void edge_expsum1(const int* __restrict__ ei, float* __restrict__ ebuf,
                             const float* __restrict__ mbuf, float* __restrict__ sbuf,
                             int E, int N) {
    long long e = (long long)blockIdx.x * blockDim.x + threadIdx.x;
    if (e >= (long long)E + N) return;
    int src, dst; edge_ids(ei, e, E, &src, &dst); (void)src;
    #pragma unroll
    for (int hh = 0; hh < 2; ++hh) {
        float t = expf(ebuf[e * 2 + hh] - mbuf[(size_t)dst * 2 + hh]);
        ebuf[e * 2 + hh] = t;
        atomicAdd(&sbuf[(size_t)dst * 2 + hh], t);
    }
}

// Edge pass 3 (layer 1): agg[dst] += h1[src] * alpha.  One wave per edge,
// lane l covers elements [4l,4l+4) of the 128-wide row; head = l>>4.
__global__ void edge_aggr1(const float* __restrict__ h1, const float* __restrict__ ebuf,
                           const float* __restrict__ sbuf, const int* __restrict__ ei,
                           float* __restrict__ agg, int E, int N) {
    long long e = (long long)blockIdx.x * (blockDim.x >> 5) + (threadIdx.x >> 5);
    if (e >= (long long)E + N) return;
    int lane = threadIdx.x & 31;
    int src, dst; edge_ids(ei, e, E, &src, &dst);
    int head = lane >> 4;
    float alpha = ebuf[e * 2 + head] / (sbuf[(size_t)dst * 2 + head] + GAT_EPS);
    float4 hv = ((const float4*)(h1 + (size_t)src * 128))[lane];
    float* orow = agg + (size_t)dst * 128 + lane * 4;
    atomicAdd(orow + 0, hv.x * alpha);
    atomicAdd(orow + 1, hv.y * alpha);
    atomicAdd(orow + 2, hv.z * alpha);
    atomicAdd(orow + 3, hv.w * alpha);
}

// x2 = elu(agg + b1), in place
__global__ void elu_bias1(float* __restrict__ agg, const float* __restrict__ b1, long long n) {
    long long i = (long long)blockIdx.x * blockDim.x + threadIdx.x;
    if (i >= n) return;
    float v = agg[i] + b1[i & 127];
    agg[i] = v > 0.0f ? v : (expf(v) - 1.0f);
}

// ---------------------------------------------------------------------------
// Layer 2: h2[n,0:2] = x2[n,:] @ W2[128,2]; plus att logits. Wave per node.
// ---------------------------------------------------------------------------
__global__ void gemm2_att2(const float* __restrict__ x2, const float* __restrict__ W2,
                           const float* __restrict__ as2, const float* __restrict__ ad2,
                           float* __restrict__ h2, float* __restrict__ a_src,
                           float* __restrict__ a_dst, int N) {
    int node = blockIdx.x * (blockDim.x >> 5) + (threadIdx.x >> 5);
    if (node >= N) return;
    int lane = threadIdx.x & 31;
    float4 xv = ((const float4*)(x2 + (size_t)node * 128))[lane];
    int k = lane * 4;
    float acc0 = xv.x * W2[(k + 0) * 2 + 0] + xv.y * W2[(k + 1) * 2 + 0] +
                 xv.z * W2[(k + 2) * 2 + 0] + xv.w * W2[(k + 3) * 2 + 0];
    float acc1 = xv.x * W2[(k + 0) * 2 + 1] + xv.y * W2[(k + 1) * 2 + 1] +
                 xv.z * W2[(k + 2) * 2 + 1] + xv.w * W2[(k + 3) * 2 + 1];
    #pragma unroll
    for (int m = 1; m <= 16; m <<= 1) {
        acc0 += __shfl_xor(acc0, m, 32);
        acc1 += __shfl_xor(acc1, m, 32);
    }
    if (lane == 0) {
        h2[(size_t)node * 2 + 0] = acc0;
        h2[(size_t)node * 2 + 1] = acc1;
        a_src[node] = acc0 * as2[0] + acc1 * as2[1];
        a_dst[node] = acc0 * ad2[0] + acc1 * ad2[1];
    }
}

// Layer-2 edge passes (H=1)
__global__ void edge_max2(const float* __restrict__ a_src, const float* __restrict__ a_dst,
                          const int* __restrict__ ei, float* __restrict__ ebuf,
                          float* __restrict__ mbuf, int E, int N) {
    long long e = (long long)blockIdx.x * blockDim.x + threadIdx.x;
    if (e >= (long long)E + N) return;
    int src, dst; edge_ids(ei, e, E, &src, &dst);
    float v = a_src[src] + a_dst[dst];
    v = v > 0.0f ? v : NEG_SLOPE * v;
    ebuf[e] = v;
    atomicMaxF32(&mbuf[dst], v);
}

__global__ void edge_expsum2(const int* __restrict__ ei, float* __restrict__ ebuf,
                             const float* __restrict__ mbuf, float* __restrict__ sbuf,
                             int E, int N) {
    long long e = (long long)blockIdx.x * blockDim.x + threadIdx.x;
    if (e >= (long long)E + N) return;
    int src, dst; edge_ids(ei, e, E, &src, &dst); (void)src;
    float t = expf(ebuf[e] - mbuf[dst]);
    ebuf[e] = t;
    atomicAdd(&sbuf[dst], t);
}

__global__ void edge_aggr2(const float* __restrict__ h2, const float* __restrict__ ebuf,
                           const float* __restrict__ sbuf, const int* __restrict__ ei,
                           float* __restrict__ out, int E, int N) {
    long long e = (long long)blockIdx.x * blockDim.x + threadIdx.x;
    if (e >= (long long)E + N) return;
    int src, dst; edge_ids(ei, e, E, &src, &dst);
    float alpha = ebuf[e] / (sbuf[dst] + GAT_EPS);
    atomicAdd(&out[(size_t)dst * 2 + 0], h2[(size_t)src * 2 + 0] * alpha);
    atomicAdd(&out[(size_t)dst * 2 + 1], h2[(size_t)src * 2 + 1] * alpha);
}

// ---------------------------------------------------------------------------
extern "C" void kernel_launch(void* const* d_in, const int* in_sizes, int n_in,
                              void* d_out, int out_size, void* d_ws, size_t ws_size,
                              hipStream_t stream) {
    const float* x   = (const float*)d_in[0];
    const int*   ei  = (const int*)  d_in[1];
    const float* W1  = (const float*)d_in[2];
    const float* as1 = (const float*)d_in[3];
    const float* ad1 = (const float*)d_in[4];
    const float* b1  = (const float*)d_in[5];
    const float* W2  = (const float*)d_in[6];
    const float* as2 = (const float*)d_in[7];
    const float* ad2 = (const float*)d_in[8];
    const float* b2  = (const float*)d_in[9];
    float* out = (float*)d_out;

    const int N = in_sizes[0] / 128;
    const int E = in_sizes[1] / 2;
    const long long TOT = (long long)E + N;

    // ---- workspace carve-out (256B aligned slabs) ----
    char* wptr = (char*)d_ws;
    auto carve = [&](size_t bytes) -> float* {
        float* p = (float*)wptr;
        wptr += (bytes + 255) & ~(size_t)255;
        return p;
    };
    float* h1     = carve((size_t)N * 128 * 4);
    float* agg1   = carve((size_t)N * 128 * 4);   // becomes x2 after elu
    float* a_src1 = carve((size_t)N * 2 * 4);
    float* a_dst1 = carve((size_t)N * 2 * 4);
    float* m1     = carve((size_t)N * 2 * 4);
    float* s1     = carve((size_t)N * 2 * 4);
    float* ebuf   = carve((size_t)TOT * 2 * 4);   // reused as layer-2 e (TOT floats)
    float* h2     = carve((size_t)N * 2 * 4);
    float* a_src2 = carve((size_t)N * 4);
    float* a_dst2 = carve((size_t)N * 4);
    float* m2     = carve((size_t)N * 4);
    float* s2     = carve((size_t)N * 4);

    const int BT = 256;
    auto blocks = [&](long long n) -> unsigned { return (unsigned)((n + BT - 1) / BT); };
    const unsigned edgeBlk  = blocks(TOT);           // thread per edge
    const unsigned edgeWBlk = blocks(TOT * 32);      // wave per edge
    const unsigned nodeWBlk = blocks((long long)N * 32);

    // ---- init ----
    fill_f32<<<blocks((long long)N * 2), BT, 0, stream>>>(m1, -INFINITY, (long long)N * 2);
    fill_f32<<<blocks((long long)N * 2), BT, 0, stream>>>(s1, 0.0f, (long long)N * 2);
    fill_f32<<<blocks((long long)N * 128), BT, 0, stream>>>(agg1, 0.0f, (long long)N * 128);
    fill_f32<<<blocks(N), BT, 0, stream>>>(m2, -INFINITY, N);
    fill_f32<<<blocks(N), BT, 0, stream>>>(s2, 0.0f, N);
    init_out_bias<<<blocks((long long)2 * N), BT, 0, stream>>>(out, b2, N);

    // ---- layer 1 ----
    gemm1_wmma<<<(N + 15) / 16, 256, 0, stream>>>(x, W1, h1, N);
    node_att1<<<nodeWBlk, BT, 0, stream>>>(h1, as1, ad1, a_src1, a_dst1, N);
    edge_max1<<<edgeBlk, BT, 0, stream>>>(a_src1, a_dst1, ei, ebuf, m1, E, N);
    edge_expsum1<<<edgeBlk, BT, 0, stream>>>(ei, ebuf, m1, s1, E, N);
    edge_aggr1<<<edgeWBlk, BT, 0, stream>>>(h1, ebuf, s1, ei, agg1, E, N);
    elu_bias1<<<blocks((long long)N * 128), BT, 0, stream>>>(agg1, b1, (long long)N * 128);

    // ---- layer 2 ----
    gemm2_att2<<<nodeWBlk, BT, 0, stream>>>(agg1, W2, as2, ad2, h2, a_src2, a_dst2, N);
    edge_max2<<<edgeBlk, BT, 0, stream>>>(a_src2, a_dst2, ei, ebuf, m2, E, N);
    edge_expsum2<<<edgeBlk, BT, 0, stream>>>(ei, ebuf, m2, s2, E, N);
    edge_aggr2<<<edgeBlk, BT, 0, stream>>>(h2, ebuf, s2, ei, out, E, N);
}